// Earth4D_27943057227892
// MI455X (gfx1250) — compile-verified
//
#include <hip/hip_runtime.h>
#include <math.h>
#include <stdint.h>

#define NSPAT 24
#define NTEMP 19
#define NLV   (NSPAT + NTEMP)
#define BLOCK 256

struct LevelSpec {
  float rx, ry, rz;          // level resolution (float, for pos = x01*res)
  uint32_t cx, cy, cz;       // clamp limit = res (uint)
  uint32_t gp0, gp01;        // dense strides: gp0, gp0*gp1
  uint32_t mask;             // hashed: size-1
  uint32_t offset;           // entry offset into table
  uint32_t hashed;           // 0 dense / 1 hashed
  uint32_t pad;
};

struct Params {
  const float4* xyzt;
  const float2* spat;
  const float2* txy;   // xyt_table
  const float2* tyz;   // yzt_table
  const float2* txz;   // xzt_table
  float2* outS;        // [N, 24] float2
  float2* outT;        // [N, 57] float2
  uint32_t N;
  uint32_t _pad;
  LevelSpec lv[NLV];
};

__device__ __forceinline__ uint32_t umin_(uint32_t a, uint32_t b) { return a < b ? a : b; }

__device__ __forceinline__ float2 enc_level(const LevelSpec s, float x, float y, float z,
                                            const float2* __restrict__ tab) {
  float px = x * s.rx, py = y * s.ry, pz = z * s.rz;
  float fx = floorf(px), fy = floorf(py), fz = floorf(pz);
  float wx1 = px - fx, wy1 = py - fy, wz1 = pz - fz;
  float wx0 = 1.0f - wx1, wy0 = 1.0f - wy1, wz0 = 1.0f - wz1;
  uint32_t ix = (uint32_t)fx, iy = (uint32_t)fy, iz = (uint32_t)fz;
  uint32_t x0 = umin_(ix, s.cx), x1 = umin_(ix + 1u, s.cx);
  uint32_t y0 = umin_(iy, s.cy), y1 = umin_(iy + 1u, s.cy);
  uint32_t z0 = umin_(iz, s.cz), z1 = umin_(iz + 1u, s.cz);
  uint32_t i000, i100, i010, i110, i001, i101, i011, i111;
  if (s.hashed) {
    const uint32_t P = 805459861u;   // reference uses this prime for BOTH dim1 and dim2
    uint32_t hy0 = y0 * P, hy1 = y1 * P, hz0 = z0 * P, hz1 = z1 * P;
    uint32_t m = s.mask;
    i000 = (x0 ^ hy0 ^ hz0) & m;  i100 = (x1 ^ hy0 ^ hz0) & m;
    i010 = (x0 ^ hy1 ^ hz0) & m;  i110 = (x1 ^ hy1 ^ hz0) & m;
    i001 = (x0 ^ hy0 ^ hz1) & m;  i101 = (x1 ^ hy0 ^ hz1) & m;
    i011 = (x0 ^ hy1 ^ hz1) & m;  i111 = (x1 ^ hy1 ^ hz1) & m;
  } else {
    uint32_t a0 = y0 * s.gp0, a1 = y1 * s.gp0;
    uint32_t b0 = z0 * s.gp01, b1 = z1 * s.gp01;
    i000 = x0 + a0 + b0;  i100 = x1 + a0 + b0;
    i010 = x0 + a1 + b0;  i110 = x1 + a1 + b0;
    i001 = x0 + a0 + b1;  i101 = x1 + a0 + b1;
    i011 = x0 + a1 + b1;  i111 = x1 + a1 + b1;
  }
  const float2* __restrict__ base = tab + s.offset;
  float2 f000 = base[i000], f100 = base[i100], f010 = base[i010], f110 = base[i110];
  float2 f001 = base[i001], f101 = base[i101], f011 = base[i011], f111 = base[i111];
  float w000 = wx0 * wy0 * wz0, w100 = wx1 * wy0 * wz0;
  float w010 = wx0 * wy1 * wz0, w110 = wx1 * wy1 * wz0;
  float w001 = wx0 * wy0 * wz1, w101 = wx1 * wy0 * wz1;
  float w011 = wx0 * wy1 * wz1, w111 = wx1 * wy1 * wz1;
  float2 acc;
  acc.x = fmaf(w111, f111.x, fmaf(w011, f011.x, fmaf(w101, f101.x, fmaf(w001, f001.x,
          fmaf(w110, f110.x, fmaf(w010, f010.x, fmaf(w100, f100.x, w000 * f000.x)))))));
  acc.y = fmaf(w111, f111.y, fmaf(w011, f011.y, fmaf(w101, f101.y, fmaf(w001, f001.y,
          fmaf(w110, f110.y, fmaf(w010, f010.y, fmaf(w100, f100.y, w000 * f000.y)))))));
  return acc;
}

__global__ __launch_bounds__(BLOCK) void earth4d_kernel(Params P) {
  __shared__ float4 s_pts[BLOCK];
  const uint32_t tid = threadIdx.x;
  const uint32_t pt = blockIdx.x * BLOCK + tid;
  if (pt >= P.N) return;

  // Stage this thread's point through the gfx1250 async-to-LDS path:
  // global_load_async_to_lds_b128 (ASYNCcnt) -> s_wait_asynccnt -> ds_load_b128.
  {
    uint32_t ldsAddr = (uint32_t)(uintptr_t)(&s_pts[tid]);
    const float4* gp4 = P.xyzt + pt;
    asm volatile("global_load_async_to_lds_b128 %0, %1, off"
                 :: "v"(ldsAddr), "v"(gp4) : "memory");
    asm volatile("s_wait_asynccnt 0x0" ::: "memory");
  }
  float4 q = s_pts[tid];

  // Normalized coords (spatial_bound = temporal_bound = 1)
  float X = fminf(fmaxf((q.x + 1.0f) * 0.5f, 0.0f), 1.0f);
  float Y = fminf(fmaxf((q.y + 1.0f) * 0.5f, 0.0f), 1.0f);
  float Z = fminf(fmaxf((q.z + 1.0f) * 0.5f, 0.0f), 1.0f);
  float ts = (q.w * 2.0f - 1.0f) * 0.9f;
  float T = fminf(fmaxf((ts + 1.0f) * 0.5f, 0.0f), 1.0f);

  // Spatial: two levels per iteration -> overlapped gathers + one b128 store.
  // rowS byte offset = pt*192, 16B aligned, so float4 stores are legal.
  float4* __restrict__ rowS4 = (float4*)(P.outS + (size_t)pt * NSPAT);
#pragma unroll 1
  for (int l = 0; l < NSPAT; l += 2) {
    float2 a = enc_level(P.lv[l],     X, Y, Z, P.spat);
    float2 b = enc_level(P.lv[l + 1], X, Y, Z, P.spat);
    float4 v; v.x = a.x; v.y = a.y; v.z = b.x; v.w = b.y;
    rowS4[l >> 1] = v;
  }

  // Temporal rows are only 8B aligned (57 float2 per row) -> float2 stores.
  float2* __restrict__ rowT = P.outT + (size_t)pt * (3 * NTEMP);
#pragma unroll 2
  for (int l = 0; l < NTEMP; ++l)
    rowT[l] = enc_level(P.lv[NSPAT + l], X, Y, T, P.txy);
#pragma unroll 2
  for (int l = 0; l < NTEMP; ++l)
    rowT[NTEMP + l] = enc_level(P.lv[NSPAT + l], Y, Z, T, P.tyz);
#pragma unroll 2
  for (int l = 0; l < NTEMP; ++l)
    rowT[2 * NTEMP + l] = enc_level(P.lv[NSPAT + l], X, Z, T, P.txz);
}

// Host-side: replicate numpy's _level_specs in double precision.
static void build_specs(double bx, double by, double bz,
                        double mx, double my, double mz,
                        int L, int log2T, LevelSpec* lv) {
  double base[3] = {bx, by, bz};
  double maxr[3] = {mx, my, mz};
  double scale[3];
  for (int d = 0; d < 3; ++d) scale[d] = exp(log(maxr[d] / base[d]) / (double)(L - 1));
  long long T = 1ll << log2T;
  long long off = 0;
  for (int l = 0; l < L; ++l) {
    long long res[3];
    for (int d = 0; d < 3; ++d) res[d] = (long long)ceil(base[d] * pow(scale[d], (double)l));
    long long gp0 = res[0] + 1, gp1 = res[1] + 1, gp2 = res[2] + 1;
    long long dense = gp0 * gp1 * gp2;
    int hashed = dense > T;
    long long size = hashed ? T : dense;
    lv[l].rx = (float)res[0]; lv[l].ry = (float)res[1]; lv[l].rz = (float)res[2];
    lv[l].cx = (uint32_t)res[0]; lv[l].cy = (uint32_t)res[1]; lv[l].cz = (uint32_t)res[2];
    lv[l].gp0 = (uint32_t)gp0;
    lv[l].gp01 = (uint32_t)(uint64_t)(gp0 * gp1);
    lv[l].mask = (uint32_t)(size - 1);
    lv[l].offset = (uint32_t)off;
    lv[l].hashed = (uint32_t)hashed;
    lv[l].pad = 0;
    off += size;
  }
}

extern "C" void kernel_launch(void* const* d_in, const int* in_sizes, int n_in,
                              void* d_out, int out_size, void* d_ws, size_t ws_size,
                              hipStream_t stream) {
  (void)n_in; (void)out_size; (void)d_ws; (void)ws_size;
  Params P;
  P.xyzt = (const float4*)d_in[0];
  P.spat = (const float2*)d_in[1];
  P.txy  = (const float2*)d_in[2];
  P.tyz  = (const float2*)d_in[3];
  P.txz  = (const float2*)d_in[4];
  uint32_t N = (uint32_t)(in_sizes[0] / 4);
  P.N = N;
  P._pad = 0;
  P.outS = (float2*)d_out;
  P.outT = (float2*)((float*)d_out + (size_t)N * 2 * NSPAT);  // temporal part after [N,48]
  build_specs(16.0, 16.0, 16.0, 134217728.0, 134217728.0, 134217728.0, NSPAT, 22, P.lv);
  build_specs(8.0, 8.0, 8.0, 32.0, 32.0, 16.0, NTEMP, 18, P.lv + NSPAT);

  dim3 grid((N + BLOCK - 1) / BLOCK);
  earth4d_kernel<<<grid, BLOCK, 0, stream>>>(P);
}